// GateRecurrent_81003083203089
// MI455X (gfx1250) — compile-verified
//
#include <hip/hip_runtime.h>
#include <stdint.h>

// ---------------------------------------------------------------------------
// GateRecurrent L->R scan, [B=8, C=32, H=256, W=256] fp32.
// Bandwidth-bound: 5 x 64 MiB of traffic, ~0.13 GFLOP -> ~14 us floor at
// 23.3 TB/s. No WMMA role (tridiagonal per-step operator, serial in W).
// CDNA5 path used: Tensor Data Mover strided-2D tile loads into LDS
// (double-buffered, TDM pad feature -> 18-dword LDS row stride, conflict-free
// column reads), synchronized with s_wait_tensorcnt + workgroup barriers.
// 1 WG per (b,c) plane: 256 WGs x 256 threads (8 wave32s), thread = H row.
// ---------------------------------------------------------------------------

#define H_DIM 256
#define W_DIM 256
#define WT    16                      // tile width (columns per TDM tile)
#define NT    (W_DIM / WT)            // 16 tiles
#define ROWSTRIDE 18                  // dwords: 16 data + 2 TDM pad dwords
#define TILE_FLOATS (H_DIM * ROWSTRIDE)       // 4608 floats / tensor / buffer
#define HB_OFF (2 * 4 * TILE_FLOATS)          // 36864: start of h-exchange buf
#define HB_LEN 258                            // [0]=zero, [1..256]=h, [257]=zero
#define SMEM_FLOATS (HB_OFF + 2 * HB_LEN)     // 37380
#define SMEM_BYTES  (SMEM_FLOATS * 4)         // 149520 B (< 320 KB WGP LDS)

typedef unsigned int uint32x4 __attribute__((ext_vector_type(4)));
typedef int          int32x4  __attribute__((ext_vector_type(4)));
typedef int          int32x8  __attribute__((ext_vector_type(8)));

#if defined(__has_builtin)
# if __has_builtin(__builtin_amdgcn_tensor_load_to_lds)
#  define HAVE_TDM 1
# else
#  define HAVE_TDM 0
# endif
#else
# define HAVE_TDM 0
#endif

__device__ __forceinline__ void wait_tensorcnt0() {
#if defined(__has_builtin) && __has_builtin(__builtin_amdgcn_s_wait_tensorcnt)
  __builtin_amdgcn_s_wait_tensorcnt(0);
#else
  asm volatile("s_wait_tensorcnt 0x0" ::: "memory");
#endif
}

#if HAVE_TDM
// Issue one TDM 2D tile load: [tile_dim1=256 rows] x [tile_dim0=16 fp32],
// tensor_dim0_stride = 256 elements (1 KiB rows), dest LDS with pad:
// pad_interval=3 (every 16 dwords) + pad_amount=1 (2 dwords) -> 18-dword rows.
__device__ __forceinline__ void tdm_load_tile(const float* gsrc,
                                              unsigned lds_byte_off) {
  const uint64_t ga = (uint64_t)(uintptr_t)gsrc;
  uint32x4 g0;
  g0.x = 1u;                                    // count=1 valid user descriptor
  g0.y = lds_byte_off;                          // lds_addr (bytes)
  g0.z = (unsigned)(ga & 0xFFFFFFFFu);          // global_addr[31:0]
  g0.w = (unsigned)((ga >> 32) & 0x01FFFFFFu)   // global_addr[56:32]
       | (2u << 30);                            // type = 2 ("image")
  int32x8 g1;
  g1[0] = (int)((2u << 16)        // data_size = 4 B
              | (1u << 20)        // pad_enable
              | (3u << 22)        // pad_interval: 16 dwords
              | (1u << 25));      // pad_amount: 2 dwords
  g1[1] = (int)(((unsigned)W_DIM & 0xFFFFu) << 16);  // tensor_dim0 lo
  g1[2] = (int)(((unsigned)H_DIM & 0xFFFFu) << 16);  // dim0 hi=0 | tensor_dim1 lo
  g1[3] = (int)(((unsigned)WT) << 16);               // dim1 hi=0 | tile_dim0
  g1[4] = H_DIM;                                     // tile_dim1 | tile_dim2=0
  g1[5] = W_DIM;                                     // tensor_dim0_stride lo
  g1[6] = 0;                                         // stride hi | dim1_stride lo
  g1[7] = 0;
  int32x4 g2 = {0, 0, 0, 0};                         // 2D tensor: groups 2/3 zero
  int32x4 g3 = {0, 0, 0, 0};
#if __clang_major__ >= 23
  int32x8 g4 = {0, 0, 0, 0, 0, 0, 0, 0};
  __builtin_amdgcn_tensor_load_to_lds(g0, g1, g2, g3, g4, 0);
#else
  __builtin_amdgcn_tensor_load_to_lds(g0, g1, g2, g3, 0);
#endif
}
#endif  // HAVE_TDM

__global__ __launch_bounds__(256) void GateRecurrent_scan_kernel(
    const float* __restrict__ X,  const float* __restrict__ G1,
    const float* __restrict__ G2, const float* __restrict__ G3,
    float* __restrict__ Out) {
  extern __shared__ float smem[];
  const int    tid   = threadIdx.x;                  // H row index
  const int    plane = blockIdx.x;                   // b*C + c
  const size_t base  = (size_t)plane * (H_DIM * W_DIM);

  const float* ins[4] = {X + base, G1 + base, G2 + base, G3 + base};

  // Zero the h-exchange boundary slots (never written by the scan).
  if (tid == 0) {
    smem[HB_OFF + 0]            = 0.0f;
    smem[HB_OFF + 257]          = 0.0f;
    smem[HB_OFF + HB_LEN + 0]   = 0.0f;
    smem[HB_OFF + HB_LEN + 257] = 0.0f;
  }

#if HAVE_TDM
  // Low 32 bits of a flat shared pointer == LDS byte offset (shared aperture).
  const unsigned lds_base = (unsigned)(uintptr_t)(void*)smem;
  if (tid == 0) {                       // one wave issues; TDM ignores EXEC
    for (int q = 0; q < 4; ++q)
      tdm_load_tile(ins[q], lds_base + (unsigned)(q * TILE_FLOATS) * 4u);
    wait_tensorcnt0();
  }
#else
  // Cooperative synchronous load of tile 0 into buffer 0.
  for (int q = 0; q < 4; ++q) {
    const float* src = ins[q] + (size_t)tid * W_DIM;
    float*       dst = &smem[q * TILE_FLOATS + tid * ROWSTRIDE];
#pragma unroll
    for (int j = 0; j < WT; ++j) dst[j] = src[j];
  }
#endif
  __syncthreads();                      // tile 0 visible to all 8 waves

  float h = 0.0f;                       // zero-init carry (matches reference)
  int   p = 0;                          // h-exchange double-buffer parity

  for (int k = 0; k < NT; ++k) {
#if HAVE_TDM
    const int buf = k & 1;
    if (tid == 0 && (k + 1) < NT) {     // async prefetch of next tile
      const unsigned nb = (unsigned)((k + 1) & 1);
      for (int q = 0; q < 4; ++q)
        tdm_load_tile(ins[q] + (size_t)(k + 1) * WT,
                      lds_base + (unsigned)((nb * 4 + q) * TILE_FLOATS) * 4u);
    }
#else
    const int buf = 0;
    if (k > 0) {
      __syncthreads();                  // all readers of previous tile done
      for (int q = 0; q < 4; ++q) {
        const float* src = ins[q] + (size_t)tid * W_DIM + (size_t)k * WT;
        float*       dst = &smem[q * TILE_FLOATS + tid * ROWSTRIDE];
#pragma unroll
        for (int j = 0; j < WT; ++j) dst[j] = src[j];
      }
      __syncthreads();
    }
#endif

    const float* bx  = &smem[(buf * 4 + 0) * TILE_FLOATS];
    const float* bg1 = &smem[(buf * 4 + 1) * TILE_FLOATS];
    const float* bg2 = &smem[(buf * 4 + 2) * TILE_FLOATS];
    const float* bg3 = &smem[(buf * 4 + 3) * TILE_FLOATS];

    float outv[WT];
#pragma unroll
    for (int tc = 0; tc < WT; ++tc) {
      const int   ro = tid * ROWSTRIDE + tc;   // stride 18 -> conflict-free
      const float x  = bx[ro];
      const float g1 = bg1[ro];
      const float g2 = bg2[ro];
      const float g3 = bg3[ro];

      float* hb = &smem[HB_OFF + p * HB_LEN];
      hb[tid + 1] = h;                  // publish carry for neighbors
      __syncthreads();
      const float hm = hb[tid];         // h_prev[h-1] (0 at h==0)
      const float hp = hb[tid + 2];     // h_prev[h+1] (0 at h==255)

      const float g = g1 + g2 + g3;
      h = (1.0f - g) * x + g1 * hm + g2 * h + g3 * hp;
      outv[tc] = h;
      p ^= 1;                           // double buffer: 1 barrier / column
    }

    // Contiguous 64 B per row per tile.
    float4* orow = (float4*)(Out + base + (size_t)tid * W_DIM + (size_t)k * WT);
#pragma unroll
    for (int j = 0; j < WT / 4; ++j) {
      float4 v;
      v.x = outv[4 * j + 0];
      v.y = outv[4 * j + 1];
      v.z = outv[4 * j + 2];
      v.w = outv[4 * j + 3];
      orow[j] = v;
    }

#if HAVE_TDM
    if (tid == 0) wait_tensorcnt0();    // prefetched tile landed in LDS
    __syncthreads();                    // publish it; old buffer now reusable
#endif
  }
}

extern "C" void kernel_launch(void* const* d_in, const int* in_sizes, int n_in,
                              void* d_out, int out_size, void* d_ws,
                              size_t ws_size, hipStream_t stream) {
  (void)in_sizes; (void)n_in; (void)out_size; (void)d_ws; (void)ws_size;
  const float* X  = (const float*)d_in[0];
  const float* G1 = (const float*)d_in[1];
  const float* G2 = (const float*)d_in[2];
  const float* G3 = (const float*)d_in[3];
  float*       Out = (float*)d_out;

  dim3 grid(8 * 32);    // one workgroup per (b,c) plane
  dim3 block(H_DIM);    // one thread per H row, 8 wave32s
  GateRecurrent_scan_kernel<<<grid, block, SMEM_BYTES, stream>>>(X, G1, G2, G3,
                                                                 Out);
}